// MultiHeadattention_64252710748306
// MI455X (gfx1250) — compile-verified
//
#include <hip/hip_runtime.h>

// ---------------------------------------------------------------------------
// MHA forward for B=1, T=4096, C=1024, H=16, D=64 (fp32) on gfx1250 (CDNA5).
// All matmuls use native fp32 WMMA: V_WMMA_F32_16X16X4_F32.
// Attention K-tiles are staged into LDS by the Tensor Data Mover (TDM).
// ---------------------------------------------------------------------------

typedef float v2f __attribute__((ext_vector_type(2)));
typedef float v8f __attribute__((ext_vector_type(8)));
typedef unsigned int u32x4 __attribute__((ext_vector_type(4)));
typedef int i32x4 __attribute__((ext_vector_type(4)));
typedef int i32x8 __attribute__((ext_vector_type(8)));

#define T_  4096
#define C_  1024
#define H_  16
#define D_  64
#define C3_ 3072

// D(16x16,f32) = A(16x4,f32) x B(4x16,f32) + C
// A frag: lane m=lane&15 holds A[m][k], A[m][k+1] with k = 2*(lane>>4)
// B frag: lane n=lane&15 holds B[k][n], B[k+1][n] with k = 2*(lane>>4)
// C/D:    vgpr i, lane: row = i + 8*(lane>>4), col = lane&15
__device__ __forceinline__ v8f wmma_f32(v2f a, v2f b, v8f c) {
  return __builtin_amdgcn_wmma_f32_16x16x4_f32(
      /*neg_a=*/false, a, /*neg_b=*/false, b,
      /*c_mod=*/(short)0, c, /*reuse_a=*/false, /*reuse_b=*/false);
}

// Butterfly reductions across 16-lane halves of the wave32.
// ds_swizzle group-of-32 encoding: [14:10]=xor, [9:5]=or, [4:0]=and.
__device__ __forceinline__ float red_max16(float v) {
  v = fmaxf(v, __int_as_float(__builtin_amdgcn_ds_swizzle(__float_as_int(v), 0x041f)));
  v = fmaxf(v, __int_as_float(__builtin_amdgcn_ds_swizzle(__float_as_int(v), 0x081f)));
  v = fmaxf(v, __int_as_float(__builtin_amdgcn_ds_swizzle(__float_as_int(v), 0x101f)));
  v = fmaxf(v, __int_as_float(__builtin_amdgcn_ds_swizzle(__float_as_int(v), 0x201f)));
  return v;
}
__device__ __forceinline__ float red_sum16(float v) {
  v += __int_as_float(__builtin_amdgcn_ds_swizzle(__float_as_int(v), 0x041f));
  v += __int_as_float(__builtin_amdgcn_ds_swizzle(__float_as_int(v), 0x081f));
  v += __int_as_float(__builtin_amdgcn_ds_swizzle(__float_as_int(v), 0x101f));
  v += __int_as_float(__builtin_amdgcn_ds_swizzle(__float_as_int(v), 0x201f));
  return v;
}

// ---------------------------------------------------------------------------
// TDM: DMA a 2D f32 tile (rows x cols, global row stride in elements) from
// global memory into LDS, padding each row of `cols` DWORDs with extra DWORDs
// (hardware pad: interval code c -> every 2^(c+1) DWORDs, amount code a ->
// a+1 DWORDs). D# layout per CDNA5 ISA 8.3/8.4 (group0 128b, group1 256b,
// groups 2/3 zero for a 2D tile; trailing 8-dword group unused -> zero).
// Must be issued by a single wave; completion via s_wait_tensorcnt.
// ---------------------------------------------------------------------------
__device__ __forceinline__ void tdm_load_2d_f32(
    unsigned ldsByteAddr, const float* gsrc,
    unsigned rows, unsigned cols, unsigned rowStrideElems,
    unsigned padIntervalCode, unsigned padAmountCode) {
  unsigned long long ga = (unsigned long long)(uintptr_t)gsrc;
  u32x4 g0;
  g0[0] = 1u;                                    // count=1, user mode, no gather
  g0[1] = ldsByteAddr;                           // lds_addr [63:32]
  g0[2] = (unsigned)(ga & 0xFFFFFFFFu);          // global_addr [95:64]
  g0[3] = (unsigned)((ga >> 32) & 0x01FFFFFFu)   // global_addr [120:96]
          | (2u << 30);                          // type=2 ("image")
  i32x8 g1;
  g1[0] = (int)((2u << 16)                       // data_size = 4 bytes
                | (1u << 20)                     // pad_enable
                | (padIntervalCode << 22)        // pad every 2^(c+1) DWORDs
                | (padAmountCode << 25));        // pad (c+1) DWORDs
  g1[1] = (int)(cols << 16);                     // tensor_dim0[15:0] (ab_addr=0)
  g1[2] = (int)((cols >> 16) | (rows << 16));    // tensor_dim0 hi | tensor_dim1 lo
  g1[3] = (int)((rows >> 16) | (cols << 16));    // tensor_dim1 hi | tile_dim0
  g1[4] = (int)rows;                             // tile_dim1 | tile_dim2=0
  g1[5] = (int)rowStrideElems;                   // tensor_dim0_stride lo32
  g1[6] = 0;                                     // stride hi | dim1_stride lo
  g1[7] = 0;                                     // dim1_stride hi
  i32x4 z4 = {0, 0, 0, 0};                       // groups 2/3: 2D tile
  i32x8 z8 = {0, 0, 0, 0, 0, 0, 0, 0};           // trailing group (unused)
  __builtin_amdgcn_tensor_load_to_lds(g0, g1, z4, z4, z8, 0);
}

// ---------------------------------------------------------------------------
// C[M,N] = A[M,K] * W[N,K]^T + bias[N]
// Block = 128 threads (4 waves, 2x2). WG tile 128x128; wave tile 64x64
// (4x4 fragments = 128 accumulator VGPRs; 16 WMMA per 8 b64 loads).
// Grid: (N/128, M/128).
// ---------------------------------------------------------------------------
__global__ __launch_bounds__(128)
void gemm_bias_wmma(const float* __restrict__ A, const float* __restrict__ W,
                    const float* __restrict__ bias, float* __restrict__ Cout,
                    int N, int K) {
  const int lane = threadIdx.x & 31;
  const int wave = threadIdx.x >> 5;
  const int lm   = lane & 15;
  const int lk   = (lane >> 4) << 1;   // k sub-offset within a 4-step
  const int hi8  = (lane >> 4) << 3;   // row offset for C-layout

  const int mBase = blockIdx.y * 128 + (wave >> 1) * 64;
  const int nBase = blockIdx.x * 128 + (wave & 1) * 64;

  v8f acc[4][4] = {};

  const float* ap = A + (size_t)(mBase + lm) * K + lk;
  const float* bp = W + (size_t)(nBase + lm) * K + lk;

#pragma unroll 2
  for (int k0 = 0; k0 < K; k0 += 4) {
    v2f af[4], bf[4];
#pragma unroll
    for (int mi = 0; mi < 4; ++mi)
      af[mi] = *(const v2f*)(ap + (size_t)(mi * 16) * K + k0);
#pragma unroll
    for (int ni = 0; ni < 4; ++ni)
      bf[ni] = *(const v2f*)(bp + (size_t)(ni * 16) * K + k0);
#pragma unroll
    for (int mi = 0; mi < 4; ++mi)
#pragma unroll
      for (int ni = 0; ni < 4; ++ni)
        acc[mi][ni] = wmma_f32(af[mi], bf[ni], acc[mi][ni]);
  }

#pragma unroll
  for (int ni = 0; ni < 4; ++ni) {
    const int col = nBase + ni * 16 + lm;
    const float bv = bias[col];
#pragma unroll
    for (int mi = 0; mi < 4; ++mi) {
#pragma unroll
      for (int i = 0; i < 8; ++i) {
        const int row = mBase + mi * 16 + i + hi8;
        Cout[(size_t)row * N + col] = acc[mi][ni][i] + bv;
      }
    }
  }
}

// ---------------------------------------------------------------------------
// Causal flash attention over qkv[T][3C] (Q|K|V concatenated per token).
// Grid: (H, T/64). Block = 128 threads; wave w owns query rows
// qb*64 + 16*w .. +15. All 4 waves stream the SAME 64-key blocks:
//   - K tile DMA'd into LDS by the TDM (hardware row padding 64->68 DWORDs)
//   - V tile staged transposed by ALU so PV fragments are ds_load_b64
// out[t][h*64+d] = softmax(QK^T/sqrt(D), causal) @ V
// ---------------------------------------------------------------------------
__global__ __launch_bounds__(128)
void attn_causal_wmma(const float* __restrict__ qkv, float* __restrict__ out) {
  __shared__ float Kl[64][68];           // [key][d]   (pad 68 -> bank stride 4)
  __shared__ float Vt[64][68];           // [d][key]   (transposed)
  __shared__ float Pb[4][16 * 68];       // per-wave P staging

  const int lane = threadIdx.x & 31;
  const int wave = threadIdx.x >> 5;
  const int lm   = lane & 15;
  const int lk   = (lane >> 4) << 1;
  const int hi8  = (lane >> 4) << 3;

  const int h     = blockIdx.x;
  const int qRow0 = blockIdx.y * 64 + wave * 16;
  const int hOff  = h * D_;

  // LDS byte offset of Kl: low 32 bits of the flat shared-aperture address.
  const unsigned ldsK = (unsigned)(uintptr_t)&Kl[0][0];

  // Preload Q A-fragments for all 16 k-steps, pre-scaled by 1/sqrt(D)=0.125
  v2f qa[16];
  {
    const float* qp = qkv + (size_t)(qRow0 + lm) * C3_ + hOff + lk;
#pragma unroll
    for (int s = 0; s < 16; ++s) {
      v2f t = *(const v2f*)(qp + 4 * s);
      qa[s].x = t.x * 0.125f;
      qa[s].y = t.y * 0.125f;
    }
  }

  float mrow[8], lrow[8];
  v8f acc[4] = {};
#pragma unroll
  for (int i = 0; i < 8; ++i) { mrow[i] = -3.0e38f; lrow[i] = 0.0f; }

  float* pb = &Pb[wave][0];
  const int nkb = blockIdx.y + 1;        // uniform across the whole WG

  for (int kb = 0; kb < nkb; ++kb) {
    const int k0 = kb << 6;

    // ---- stage K via TDM, V transposed via ALU -> LDS ----
    __syncthreads();                     // previous block's readers are done
    {
      const float* Kg = qkv + (size_t)k0 * C3_ + C_ + hOff;
      const float* Vg = Kg + C_;
      if (wave == 0) {
        // 64x64 f32 tile, row stride 3072 elems; pad each 64-DWORD row by 4
        // DWORDs (interval code 5 = 64 DW, amount code 3 = 4 DW) -> Kl[64][68]
        tdm_load_2d_f32(ldsK, Kg, 64, 64, C3_, 5, 3);
      }
#pragma unroll
      for (int it = 0; it < 16; ++it) {  // V: 64x64 floats, b64 per thread
        const int idx = it * 128 + threadIdx.x;
        const int r   = idx >> 5;        // key row
        const int c2  = (idx & 31) << 1; // d column (even)
        v2f vv = *(const v2f*)(Vg + (size_t)r * C3_ + c2);
        Vt[c2][r]     = vv.x;
        Vt[c2 + 1][r] = vv.y;
      }
      if (wave == 0) __builtin_amdgcn_s_wait_tensorcnt(0);
    }
    __syncthreads();                     // publish Kl + Vt to all waves

    // ---- S = (Q/sqrt(D)) * K^T : 4 column tiles x 16 k-steps = 64 WMMA ----
    v8f sc[4] = {};
#pragma unroll 4
    for (int s = 0; s < 16; ++s) {
#pragma unroll
      for (int nt = 0; nt < 4; ++nt) {
        v2f bk = *(const v2f*)&Kl[nt * 16 + lm][4 * s + lk];
        sc[nt] = wmma_f32(qa[s], bk, sc[nt]);
      }
    }

    // ---- causal mask + online softmax (per row i held by this lane) ----
#pragma unroll
    for (int i = 0; i < 8; ++i) {
      const int row = qRow0 + i + hi8;
      float mx = -3.0e38f;
#pragma unroll
      for (int nt = 0; nt < 4; ++nt) {
        const int key = k0 + nt * 16 + lm;
        float vv = (key > row) ? -3.0e38f : sc[nt][i];
        sc[nt][i] = vv;
        mx = fmaxf(mx, vv);
      }
      mx = red_max16(mx);
      const float mn   = fmaxf(mrow[i], mx);
      const float corr = __expf(mrow[i] - mn);
      mrow[i] = mn;
      lrow[i] *= corr;
#pragma unroll
      for (int dt = 0; dt < 4; ++dt) acc[dt][i] *= corr;
      float rs = 0.0f;
#pragma unroll
      for (int nt = 0; nt < 4; ++nt) {
        const float p = __expf(sc[nt][i] - mn);
        sc[nt][i] = p;
        rs += p;
      }
      lrow[i] += red_sum16(rs);
    }

    // ---- transpose P through per-wave LDS (C-layout -> A-fragment) ----
#pragma unroll
    for (int nt = 0; nt < 4; ++nt)
#pragma unroll
      for (int i = 0; i < 8; ++i)
        pb[(i + hi8) * 68 + nt * 16 + lm] = sc[nt][i];
    asm volatile("s_wait_dscnt 0x0" ::: "memory");   // in-wave LDS RAW

    // ---- O += P * V : 4 d-tiles x 16 k-steps = 64 WMMA (all from LDS) ----
#pragma unroll 4
    for (int s = 0; s < 16; ++s) {
      v2f pa = *(const v2f*)(pb + lm * 68 + 4 * s + lk);
#pragma unroll
      for (int dt = 0; dt < 4; ++dt) {
        v2f bv = *(const v2f*)&Vt[dt * 16 + lm][4 * s + lk];
        acc[dt] = wmma_f32(pa, bv, acc[dt]);
      }
    }
  }

  // ---- epilogue: normalize by row sums, write (T, C) head-concat layout ----
#pragma unroll
  for (int i = 0; i < 8; ++i) {
    const float inv = 1.0f / lrow[i];
    const int row = qRow0 + i + hi8;
#pragma unroll
    for (int dt = 0; dt < 4; ++dt)
      out[(size_t)row * C_ + hOff + dt * 16 + lm] = acc[dt][i] * inv;
  }
}

// ---------------------------------------------------------------------------
extern "C" void kernel_launch(void* const* d_in, const int* in_sizes, int n_in,
                              void* d_out, int out_size, void* d_ws, size_t ws_size,
                              hipStream_t stream) {
  const float* x     = (const float*)d_in[0];
  // d_in[1] = causal mask (int32) — causality is computed analytically, unused.
  const float* Wqkv  = (const float*)d_in[2];
  const float* bqkv  = (const float*)d_in[3];
  const float* Wproj = (const float*)d_in[4];
  const float* bproj = (const float*)d_in[5];
  float* out = (float*)d_out;

  float* qkv  = (float*)d_ws;                      // T x 3C = 48 MB
  float* attn = qkv + (size_t)T_ * C3_;            // T x C  = 16 MB

  dim3 blk(128);

  // 1) qkv = x @ W_qkv^T + b_qkv        (4096 x 3072)
  gemm_bias_wmma<<<dim3(C3_ / 128, T_ / 128), blk, 0, stream>>>(
      x, Wqkv, bqkv, qkv, C3_, C_);

  // 2) attn = causal_softmax(Q K^T / sqrt(D)) @ V   (per head, flash style)
  attn_causal_wmma<<<dim3(H_, T_ / 64), blk, 0, stream>>>(qkv, attn);

  // 3) out = attn @ W_proj^T + b_proj   (4096 x 1024)
  gemm_bias_wmma<<<dim3(C_ / 128, T_ / 128), blk, 0, stream>>>(
      attn, Wproj, bproj, out, C_, C_);
}